// DynamicKRouter_2843268350418
// MI455X (gfx1250) — compile-verified
//
#include <hip/hip_runtime.h>

#define HIDDEN   4096
#define NEXP     64
#define TOK_WG   16          // tokens per workgroup (one 16-row M tile)
#define THRESH   0.8f

typedef float v2f __attribute__((ext_vector_type(2)));
typedef float v8f __attribute__((ext_vector_type(8)));

// D = A(16x4 f32) * B(4x16 f32) + C(16x16 f32), wave32 WMMA.
// A layout: lanes 0-15 -> M=0..15, VGPR0=K0/K2, VGPR1=K1/K3 (K2/K3 in lanes 16-31)
// B layout: lanes 0-15 -> N=0..15, VGPR0=K0/K2, VGPR1=K1/K3
// C layout: VGPR i -> M = i + 8*(lane>=16), N = lane&15
__global__ __launch_bounds__(128)
void DynamicKRouter_kernel(const float* __restrict__ x,
                           const float* __restrict__ W,
                           const float* __restrict__ bias,
                           float* __restrict__ out,
                           int ntok)
{
    __shared__ float lg[TOK_WG][NEXP];   // logits, then probs
    __shared__ float sp[TOK_WG][NEXP];   // sorted probs (descending)
    __shared__ int   od[TOK_WG][NEXP];   // argsort order
    __shared__ int   kv[TOK_WG];         // per-token k

    const int tid   = threadIdx.x;
    const int wave  = tid >> 5;          // 0..3 -> expert tile of 16
    const int lane  = tid & 31;
    const int row   = lane & 15;         // M row for A, N col for B/C
    const int kh    = lane >> 4;         // K-half (0: K%4 in {0,1}, 1: {2,3})
    const int tok0  = blockIdx.x * TOK_WG;
    const int ntile = wave * 16;

    const float* __restrict__ xrow = x + (size_t)(tok0 + row) * HIDDEN;

    v8f c = {0.f, 0.f, 0.f, 0.f, 0.f, 0.f, 0.f, 0.f};

    // ---- Phase 1: GEMM, logits tile 16x16 per wave -------------------------
    for (int k = 0; k < HIDDEN; k += 16) {
        __builtin_prefetch(xrow + k + 256, 0, 0);   // global_prefetch_b8
#pragma unroll
        for (int kk = 0; kk < 16; kk += 4) {
            const int kb = k + kk;
            // A fragment: this lane holds K = kb+2*kh, kb+2*kh+1 for row M=row
            v2f a = *(const v2f*)(xrow + kb + 2 * kh);
            // B fragment: W[K][ntile+row], K = kb+2*kh, kb+2*kh+1
            v2f bb;
            bb.x = W[(size_t)(kb + 2 * kh)     * NEXP + ntile + row];
            bb.y = W[(size_t)(kb + 2 * kh + 1) * NEXP + ntile + row];
            c = __builtin_amdgcn_wmma_f32_16x16x4_f32(
                    false, a, false, bb, (short)0, c, false, false);
        }
    }

    // bias add + spill C tile to LDS in [token][expert] layout
    {
        const float bv = bias[ntile + row];
#pragma unroll
        for (int i = 0; i < 8; ++i)
            lg[i + 8 * kh][ntile + row] = c[i] + bv;
    }
    __syncthreads();

    // ---- Phase 2: softmax per token (one thread per token) -----------------
    if (tid < TOK_WG) {
        float m = lg[tid][0];
#pragma unroll 8
        for (int e = 1; e < NEXP; ++e) m = fmaxf(m, lg[tid][e]);
        float s = 0.f;
#pragma unroll 8
        for (int e = 0; e < NEXP; ++e) {
            float v = __expf(lg[tid][e] - m);
            lg[tid][e] = v;
            s += v;
        }
        const float inv = 1.f / s;
#pragma unroll 8
        for (int e = 0; e < NEXP; ++e) lg[tid][e] *= inv;
    }
    __syncthreads();

    // ---- Phase 3: stable descending rank sort ------------------------------
    // rank(e) = #{ j : p[j] > p[e]  or  (p[j]==p[e] and j<e) }  (unique ranks)
    for (int idx = tid; idx < TOK_WG * NEXP; idx += 128) {
        const int t = idx >> 6;
        const int e = idx & 63;
        const float p = lg[t][e];
        int rank = 0;
#pragma unroll 8
        for (int j = 0; j < NEXP; ++j) {
            const float pj = lg[t][j];
            rank += (pj > p) || (pj == p && j < e);
        }
        sp[t][rank] = p;
        od[t][rank] = e;
    }
    __syncthreads();

    // ---- Phase 4: cumsum + first threshold crossing ------------------------
    if (tid < TOK_WG) {
        float cum = 0.f;
        int k = NEXP;                      // "never crossed" fallback
        for (int s = 0; s < NEXP; ++s) {
            cum += sp[tid][s];
            if (cum >= THRESH) { k = s + 1; break; }
        }
        kv[tid] = k;
    }
    __syncthreads();

    // ---- Phase 5: write outputs (concatenated flat, all as float) ----------
    // [0, ntok*64)            selected_experts  (int values as float, -1 pad)
    // [ntok*64, 2*ntok*64)    selected_probs    (0 pad)
    // [2*ntok*64, +ntok)      k_values
    const size_t SPOFF = (size_t)ntok * NEXP;
    const size_t KVOFF = SPOFF * 2;
    for (int idx = tid; idx < TOK_WG * NEXP; idx += 128) {
        const int t = idx >> 6;
        const int s = idx & 63;
        const size_t g = (size_t)(tok0 + t) * NEXP + s;
        const int k = kv[t];
        out[g]         = (s < k) ? (float)od[t][s] : -1.f;
        out[SPOFF + g] = (s < k) ? sp[t][s] : 0.f;
    }
    if (tid < TOK_WG)
        out[KVOFF + tok0 + tid] = (float)kv[tid];
}

extern "C" void kernel_launch(void* const* d_in, const int* in_sizes, int n_in,
                              void* d_out, int out_size, void* d_ws, size_t ws_size,
                              hipStream_t stream) {
    const float* x    = (const float*)d_in[0];   // [4, 4096, 4096] f32
    const float* W    = (const float*)d_in[1];   // [4096, 64] f32
    const float* bias = (const float*)d_in[2];   // [64] f32
    float* out = (float*)d_out;

    const int ntok = in_sizes[0] / HIDDEN;       // 16384
    const int grid = ntok / TOK_WG;              // 1024 workgroups

    DynamicKRouter_kernel<<<grid, 128, 0, stream>>>(x, W, bias, out, ntok);
}